// RGCN_NC_63075889709117
// MI455X (gfx1250) — compile-verified
//
#include <hip/hip_runtime.h>

typedef __attribute__((ext_vector_type(2))) float v2f;
typedef __attribute__((ext_vector_type(8))) float v8f;

// ---------------------------------------------------------------- utilities
__global__ void k_zero(float* __restrict__ p, int n) {
    int i = blockIdx.x * blockDim.x + threadIdx.x;
    if (i < n) p[i] = 0.0f;
}

__global__ void k_deg(const int* __restrict__ dst, float* __restrict__ deg,
                      int total, int E, int N) {
    int i = blockIdx.x * blockDim.x + threadIdx.x;
    if (i >= total) return;
    int r = i / E;
    atomicAdd(&deg[(size_t)r * N + dst[i]], 1.0f);
}

__global__ void k_invdeg(float* __restrict__ deg, int n) {
    int i = blockIdx.x * blockDim.x + threadIdx.x;
    if (i < n) deg[i] = 1.0f / fmaxf(deg[i], 1.0f);
}

// y = relu(x + bias[d]) elementwise over [N, D]
__global__ void k_bias_relu(const float* __restrict__ x, const float* __restrict__ bias,
                            float* __restrict__ y, int n, int D) {
    int i = blockIdx.x * blockDim.x + threadIdx.x;
    if (i >= n) return;
    float v = x[i] + bias[i % D];
    y[i] = fmaxf(v, 0.0f);
}

// out[n, c] = bias[c]
__global__ void k_bias_init(float* __restrict__ out, const float* __restrict__ bias,
                            int n, int C) {
    int i = blockIdx.x * blockDim.x + threadIdx.x;
    if (i < n) out[i] = bias[i % C];
}

// ------------------------------------------------- normalized scatter-add
// acc[dst] += x[src] * invdeg[r][dst]      (D == 256, one wave32 per edge,
// each lane owns 8 consecutive floats -> two float4 loads + 8 f32 atomics,
// all of which stay resident in the 192MB L2 for this problem size)
__global__ void k_scatter(const float* __restrict__ x,
                          const int* __restrict__ src,
                          const int* __restrict__ dst,
                          const float* __restrict__ invdeg,   // [R or 1, N]
                          float* __restrict__ acc,
                          int nEdges, int N, int Eper) {
    int t    = blockIdx.x * blockDim.x + threadIdx.x;
    int edge = t >> 5;
    int lane = t & 31;
    if (edge >= nEdges) return;
    int r = edge / Eper;
    int s = src[edge];
    int d = dst[edge];
    float w = invdeg[(size_t)r * N + d];

    const float4* xs = (const float4*)(x + (size_t)s * 256 + lane * 8);
    float4 a = xs[0];
    float4 b = xs[1];
    float* ap = acc + (size_t)d * 256 + lane * 8;
    atomicAdd(ap + 0, a.x * w);
    atomicAdd(ap + 1, a.y * w);
    atomicAdd(ap + 2, a.z * w);
    atomicAdd(ap + 3, a.w * w);
    atomicAdd(ap + 4, b.x * w);
    atomicAdd(ap + 5, b.y * w);
    atomicAdd(ap + 6, b.z * w);
    atomicAdd(ap + 7, b.w * w);
}

// ------------------------------------------------------- f32 WMMA GEMM
// Cacc[M, CO] += A[M, K] * B[K, CO], one wave per 16x16 output tile,
// K-loop of V_WMMA_F32_16X16X4_F32. M % 16 == 0, CO % 16 == 0, K % 4 == 0.
//
// Fragment layout (ISA 7.12.2, wave32):
//   A 16x4 : lane<16 -> M=lane,     K = {0,1}; lane>=16 -> M=lane-16, K = {2,3}
//   B 4x16 : lane<16 -> N=lane,     K = {0,1}; lane>=16 -> N=lane-16, K = {2,3}
//   C 16x16: vgpr v  -> M = v + 8*(lane>=16), N = lane & 15
__global__ void k_gemm_wmma_f32(const float* __restrict__ A,
                                const float* __restrict__ B,
                                float* __restrict__ Cacc,
                                int M, int K, int CO) {
    int wave = (blockIdx.x * blockDim.x + threadIdx.x) >> 5;
    int lane = threadIdx.x & 31;
    int tilesN = CO >> 4;
    int tiles  = (M >> 4) * tilesN;
    if (wave >= tiles) return;                 // uniform per wave: EXEC all-1s at WMMA

    int m0   = (wave / tilesN) << 4;
    int n0   = (wave % tilesN) << 4;
    int half = lane >> 4;                      // 0: K/M-low half, 1: high half
    int ml   = lane & 15;

    // load current C tile (accumulation across sequential relation launches)
    v8f c;
    const float* crd = Cacc + (size_t)(m0 + 8 * half) * CO + n0 + ml;
#pragma unroll
    for (int v = 0; v < 8; ++v) c[v] = crd[(size_t)v * CO];

    const float* arow = A + (size_t)(m0 + ml) * K + 2 * half;   // row-major A
    const float* bcol = B + (size_t)(2 * half) * CO + n0 + ml;  // row-major B
#pragma unroll 4
    for (int k = 0; k < K; k += 4) {
        v2f a, b;
        a.x = arow[k];
        a.y = arow[k + 1];
        b.x = bcol[(size_t)k * CO];
        b.y = bcol[(size_t)(k + 1) * CO];
        c = __builtin_amdgcn_wmma_f32_16x16x4_f32(
                /*neg_a=*/false, a, /*neg_b=*/false, b,
                /*c_mod=*/(short)0, c, /*reuse_a=*/false, /*reuse_b=*/false);
    }

    float* cwr = Cacc + (size_t)(m0 + 8 * half) * CO + n0 + ml;
#pragma unroll
    for (int v = 0; v < 8; ++v) cwr[(size_t)v * CO] = c[v];
}

// ---------------------------------------------------------------- driver
static inline int nblk(long n, int t) { return (int)((n + t - 1) / t); }

extern "C" void kernel_launch(void* const* d_in, const int* in_sizes, int n_in,
                              void* d_out, int out_size, void* d_ws, size_t ws_size,
                              hipStream_t stream) {
    const float* embed = (const float*)d_in[0];
    const int*   src   = (const int*)d_in[1];
    const int*   dst   = (const int*)d_in[2];
    const float* b0    = (const float*)d_in[3];
    const float* w1    = (const float*)d_in[4];
    const float* b1    = (const float*)d_in[5];
    const float* w_out = (const float*)d_in[6];
    const float* b_out = (const float*)d_in[7];
    float*       out   = (float*)d_out;

    const int D  = in_sizes[3];              // 256
    const int C  = in_sizes[7];              // 64
    const int N  = in_sizes[0] / D;          // 50000
    const int R  = in_sizes[4] / (D * D);    // 4
    const int RE = in_sizes[1];              // R*E
    const int E  = RE / R;

    char* ws = (char*)d_ws;
    float* invdeg = (float*)ws; ws += (size_t)R * N * sizeof(float);
    float* h      = (float*)ws; ws += (size_t)N * D * sizeof(float);
    float* agg    = (float*)ws; ws += (size_t)N * D * sizeof(float);
    float* h2     = (float*)ws; ws += (size_t)N * D * sizeof(float);

    const int ND = N * D;

    // ---- in-degrees per relation -> invdeg (in place)
    k_zero  <<<nblk((long)R * N, 256), 256, 0, stream>>>(invdeg, R * N);
    k_deg   <<<nblk(RE, 256),          256, 0, stream>>>(dst, invdeg, RE, E, N);
    k_invdeg<<<nblk((long)R * N, 256), 256, 0, stream>>>(invdeg, R * N);

    // ---- layer 0: h = relu(sum_r normagg_r(embed) + b0)  (single fused buffer)
    k_zero   <<<nblk(ND, 256), 256, 0, stream>>>(agg, ND);
    k_scatter<<<nblk((long)RE * 32, 256), 256, 0, stream>>>(
        embed, src, dst, invdeg, agg, RE, N, E);
    k_bias_relu<<<nblk(ND, 256), 256, 0, stream>>>(agg, b0, h, ND, D);

    // ---- layer 1: h2 = relu(sum_r normagg_r(h) @ w1[r] + b1)
    k_zero<<<nblk(ND, 256), 256, 0, stream>>>(h2, ND);
    for (int r = 0; r < R; ++r) {
        k_zero   <<<nblk(ND, 256), 256, 0, stream>>>(agg, ND);
        k_scatter<<<nblk((long)E * 32, 256), 256, 0, stream>>>(
            h, src + (size_t)r * E, dst + (size_t)r * E,
            invdeg + (size_t)r * N, agg, E, N, E);
        int tiles = (N / 16) * (D / 16);
        k_gemm_wmma_f32<<<nblk((long)tiles * 32, 256), 256, 0, stream>>>(
            agg, w1 + (size_t)r * D * D, h2, N, D, D);
    }
    k_bias_relu<<<nblk(ND, 256), 256, 0, stream>>>(h2, b1, h2, ND, D);

    // ---- layer 2: out = sum_r normagg_r(h2) @ w_out[r] + b_out
    k_bias_init<<<nblk((long)N * C, 256), 256, 0, stream>>>(out, b_out, N * C, C);
    for (int r = 0; r < R; ++r) {
        k_zero   <<<nblk(ND, 256), 256, 0, stream>>>(agg, ND);
        k_scatter<<<nblk((long)E * 32, 256), 256, 0, stream>>>(
            h2, src + (size_t)r * E, dst + (size_t)r * E,
            invdeg + (size_t)r * N, agg, E, N, E);
        int tiles = (N / 16) * (C / 16);
        k_gemm_wmma_f32<<<nblk((long)tiles * 32, 256), 256, 0, stream>>>(
            agg, w_out + (size_t)r * D * C, out, N, D, C);
    }
}